// CCPL_28621662061093
// MI455X (gfx1250) — compile-verified
//
#include <hip/hip_runtime.h>
#include <math.h>

// Problem constants (match reference)
#define B_    8
#define C_    64
#define H_    512
#define W_    512
#define NUMS_ 256
#define EPS_  1e-7f

typedef float v2f __attribute__((ext_vector_type(2)));
typedef float v8f __attribute__((ext_vector_type(8)));

// Stage 1: one block per (batch, sample). Gather 3x3 windows for q and k,
// form per-channel diffs in LDS, compute per-neighbor squared L2 norms over
// the 64 channels as diag(D^T D) with V_WMMA_F32_16X16X4_F32 (q cols 0-7,
// k cols 8-15 packed into one 16-col matrix), then accumulate
// |dq/(|dq|+eps) - dk/(|dk|+eps)| and emit one deterministic partial sum.
__global__ __launch_bounds__(256) void ccpl_stage1(
    const float* __restrict__ fq, const float* __restrict__ fk,
    const int* __restrict__ ids, float* __restrict__ partial)
{
    __shared__ float s_dq[C_ * 8];    // D_q[c][n]
    __shared__ float s_dk[C_ * 8];    // D_k[c][n]
    __shared__ float s_gram[8 * 32];  // 16x16 Gram in C/D VGPR layout
    __shared__ float s_invq[8];
    __shared__ float s_invk[8];
    __shared__ float s_wsum[8];

    const int bs = blockIdx.x;            // 0 .. B*NUM_S-1
    const int b  = bs >> 8;               // NUM_S == 256
    const int s  = bs & (NUMS_ - 1);
    const int hi = ids[s * 2 + 0];
    const int wi = ids[s * 2 + 1];
    const int t  = threadIdx.x;

    const size_t base_b  = (size_t)b * C_ * H_ * W_;
    const size_t ctr_off = (size_t)(hi + 1) * W_ + (wi + 1);

    // ---- Phase A: gather + diff into LDS (i = c*8 + n, 2 per thread) ----
    float dqv[2], dkv[2];
#pragma unroll
    for (int r = 0; r < 2; ++r) {
        const int i  = t + r * 256;
        const int c  = i >> 3;
        const int n  = i & 7;
        const int i9 = n + (n >= 4 ? 1 : 0);   // skip center of 3x3
        const int dh = i9 / 3;
        const int dw = i9 - dh * 3;
        const size_t chan = base_b + (size_t)c * (H_ * W_);
        const size_t off  = chan + (size_t)(hi + dh) * W_ + (wi + dw);
        const size_t coff = chan + ctr_off;
        const float dq = fq[off] - fq[coff];
        const float dk = fk[off] - fk[coff];
        s_dq[i] = dq;
        s_dk[i] = dk;
        dqv[r] = dq;
        dkv[r] = dk;
    }
    __syncthreads();

    // ---- Phase B: wave 0 computes 16x16 Gram = D^T D via WMMA f32 16x16x4.
    // Column j<8 -> q neighbor j ; column j>=8 -> k neighbor j-8.
    // A (16x4, f32): lane m=lane&15, VGPR0/1 = K pair, half-waves hold K 0-1 / 2-3.
    // B (4x16) row-striped layout carries the same per-lane data for B = A^T,
    // so one float2 per lane serves as both operands.
    if (t < 32) {
        const int lane = t;
        const int m    = lane & 15;
        const int kb   = (lane >> 4) << 1;           // 0 or 2
        const float* colbase = (m < 8) ? s_dq : s_dk;
        const int n = m & 7;
        v8f acc = {};
#pragma unroll
        for (int c0 = 0; c0 < C_; c0 += 4) {
            v2f a;
            a.x = colbase[(c0 + kb    ) * 8 + n];
            a.y = colbase[(c0 + kb + 1) * 8 + n];
            acc = __builtin_amdgcn_wmma_f32_16x16x4_f32(
                false, a, false, a, (short)0, acc, false, false);
        }
#pragma unroll
        for (int r = 0; r < 8; ++r) s_gram[r * 32 + lane] = acc[r];
    }
    __syncthreads();

    // Diagonal extraction per C/D layout: G[j][j] at (VGPR j, lane j) for j<8,
    // and (VGPR j-8, lane j+16) for j>=8.
    if (t < 8) {
        const float q2 = s_gram[t * 32 + t];                 // diag j = t
        const int   j  = t + 8;
        const float k2 = s_gram[(j - 8) * 32 + (j + 16)];    // diag j = t+8
        s_invq[t] = 1.0f / (sqrtf(q2) + EPS_);
        s_invk[t] = 1.0f / (sqrtf(k2) + EPS_);
    }
    __syncthreads();

    // ---- Phase C: per-element |normalized q - normalized k|, block reduce ----
    float acc = 0.0f;
#pragma unroll
    for (int r = 0; r < 2; ++r) {
        const int n = (t + r * 256) & 7;
        acc += fabsf(dqv[r] * s_invq[n] - dkv[r] * s_invk[n]);
    }
    for (int off = 16; off > 0; off >>= 1)
        acc += __shfl_down(acc, off, 32);
    if ((t & 31) == 0) s_wsum[t >> 5] = acc;
    __syncthreads();
    if (t == 0) {
        float tot = 0.0f;
#pragma unroll
        for (int w = 0; w < 8; ++w) tot += s_wsum[w];
        partial[bs] = tot;
    }
}

// Stage 2: deterministic reduction of the 2048 per-block partials -> mean.
__global__ __launch_bounds__(256) void ccpl_stage2(
    const float* __restrict__ partial, float* __restrict__ out,
    int n, float scale)
{
    __shared__ float s_wsum[8];
    const int t = threadIdx.x;
    float acc = 0.0f;
    for (int i = t; i < n; i += 256) acc += partial[i];
    for (int off = 16; off > 0; off >>= 1)
        acc += __shfl_down(acc, off, 32);
    if ((t & 31) == 0) s_wsum[t >> 5] = acc;
    __syncthreads();
    if (t == 0) {
        float tot = 0.0f;
#pragma unroll
        for (int w = 0; w < 8; ++w) tot += s_wsum[w];
        out[0] = tot * scale;
    }
}

extern "C" void kernel_launch(void* const* d_in, const int* in_sizes, int n_in,
                              void* d_out, int out_size, void* d_ws, size_t ws_size,
                              hipStream_t stream) {
    const float* fq  = (const float*)d_in[0];
    const float* fk  = (const float*)d_in[1];
    const int*   ids = (const int*)d_in[2];
    float* out = (float*)d_out;
    float* partial = (float*)d_ws;   // B_*NUMS_ floats = 8 KB scratch

    const int nblocks = B_ * NUMS_;  // 2048
    ccpl_stage1<<<nblocks, 256, 0, stream>>>(fq, fk, ids, partial);

    const float scale = 1.0f / (float)(B_ * C_ * NUMS_ * 8);  // 1/1048576
    ccpl_stage2<<<1, 256, 0, stream>>>(partial, out, nblocks, scale);
}